// GS3DRenderer_46162308497762
// MI455X (gfx1250) — compile-verified
//
#include <hip/hip_runtime.h>
#include <hip/hip_bf16.h>
#include <math.h>

// ---------------- problem constants (fixed by the reference) ----------------
#define N_GAUSS   2048
#define IMG       96
#define NPIX      (IMG * IMG)
#define BLUR      0.3f
#define ALPHA_MAX 0.99f
#define ALPHA_MIN (1.0f / 255.0f)
#define ZNEAR     0.01f
#define GSTRIDE   12   // floats per gaussian record (3 x float4)

typedef float v2f __attribute__((ext_vector_type(2)));
typedef float v8f __attribute__((ext_vector_type(8)));

__device__ __forceinline__ float sigmoidf(float x) {
  return 1.0f / (1.0f + __expf(-x));
}

// ---------------------------------------------------------------------------
// Kernel 1: per-gaussian preprocessing.
// Record layout (12 floats): [u, v, conic_a, conic_b, conic_c, opac, r, g, b, z, 0, 0]
// ---------------------------------------------------------------------------
__global__ void gs_preprocess(const float* __restrict__ xyz,
                              const float* __restrict__ scaling,
                              const float* __restrict__ rotation,
                              const float* __restrict__ opacity,
                              const float* __restrict__ rgb,
                              const float* __restrict__ c2w,
                              const float* __restrict__ intr,
                              float* __restrict__ rec,   // [N_GAUSS * GSTRIDE]
                              float* __restrict__ zkey)  // [N_GAUSS]
{
  int gid = blockIdx.x * blockDim.x + threadIdx.x;
  if (gid >= N_GAUSS) return;

  // w2c = inverse of rigid c2w:  Rw = Rc^T,  tw = -Rc^T * tc
  float Rw[3][3], tw[3];
  for (int i = 0; i < 3; ++i)
    for (int j = 0; j < 3; ++j)
      Rw[i][j] = c2w[j * 4 + i];
  for (int i = 0; i < 3; ++i)
    tw[i] = -(Rw[i][0] * c2w[0 * 4 + 3] + Rw[i][1] * c2w[1 * 4 + 3] + Rw[i][2] * c2w[2 * 4 + 3]);

  const float fx = intr[0], cx = intr[2], fy = intr[4], cy = intr[5];

  float p0 = xyz[gid * 3 + 0], p1 = xyz[gid * 3 + 1], p2 = xyz[gid * 3 + 2];
  float x = Rw[0][0] * p0 + Rw[0][1] * p1 + Rw[0][2] * p2 + tw[0];
  float y = Rw[1][0] * p0 + Rw[1][1] * p1 + Rw[1][2] * p2 + tw[1];
  float z = Rw[2][0] * p0 + Rw[2][1] * p1 + Rw[2][2] * p2 + tw[2];
  float zc = fmaxf(z, 1e-6f);
  float u = fx * x / zc + cx;
  float v = fy * y / zc + cy;

  // quaternion -> rotation
  float qw = rotation[gid * 4 + 0], qx = rotation[gid * 4 + 1];
  float qy = rotation[gid * 4 + 2], qz = rotation[gid * 4 + 3];
  float qn = rsqrtf(qw * qw + qx * qx + qy * qy + qz * qz);
  qw *= qn; qx *= qn; qy *= qn; qz *= qn;
  float Rq[3][3] = {
    {1.f - 2.f * (qy * qy + qz * qz), 2.f * (qx * qy - qw * qz),       2.f * (qx * qz + qw * qy)},
    {2.f * (qx * qy + qw * qz),       1.f - 2.f * (qx * qx + qz * qz), 2.f * (qy * qz - qw * qx)},
    {2.f * (qx * qz - qw * qy),       2.f * (qy * qz + qw * qx),       1.f - 2.f * (qx * qx + qy * qy)}
  };
  float s0 = __expf(scaling[gid * 3 + 0]);
  float s1 = __expf(scaling[gid * 3 + 1]);
  float s2 = __expf(scaling[gid * 3 + 2]);

  // M = Rq * diag(s); Sigma = M M^T; Sc = Rw Sigma Rw^T
  float M[3][3];
  for (int i = 0; i < 3; ++i) { M[i][0] = Rq[i][0] * s0; M[i][1] = Rq[i][1] * s1; M[i][2] = Rq[i][2] * s2; }
  float Sig[3][3];
  for (int i = 0; i < 3; ++i)
    for (int k = 0; k < 3; ++k)
      Sig[i][k] = M[i][0] * M[k][0] + M[i][1] * M[k][1] + M[i][2] * M[k][2];
  float RS[3][3];
  for (int i = 0; i < 3; ++i)
    for (int k = 0; k < 3; ++k)
      RS[i][k] = Rw[i][0] * Sig[0][k] + Rw[i][1] * Sig[1][k] + Rw[i][2] * Sig[2][k];
  float Sc[3][3];
  for (int i = 0; i < 3; ++i)
    for (int l = 0; l < 3; ++l)
      Sc[i][l] = RS[i][0] * Rw[l][0] + RS[i][1] * Rw[l][1] + RS[i][2] * Rw[l][2];

  // J (2x3) projection Jacobian; cov2d = J Sc J^T
  float j00 = fx / zc, j02 = -fx * x / (zc * zc);
  float j11 = fy / zc, j12 = -fy * y / (zc * zc);
  // row0 of J*Sc
  float t00 = j00 * Sc[0][0] + j02 * Sc[2][0];
  float t01 = j00 * Sc[0][1] + j02 * Sc[2][1];
  float t02 = j00 * Sc[0][2] + j02 * Sc[2][2];
  float t10 = j11 * Sc[1][0] + j12 * Sc[2][0];
  float t11 = j11 * Sc[1][1] + j12 * Sc[2][1];
  float t12 = j11 * Sc[1][2] + j12 * Sc[2][2];
  float a = t00 * j00 + t02 * j02 + BLUR;
  float b = t00 * 0.f + t01 * j11 + t02 * j12;   // cov2d[0][1]
  float c = t10 * 0.f + t11 * j11 + t12 * j12;   // cov2d[1][1]
  (void)t10;
  c += BLUR;
  float det = fmaxf(a * c - b * b, 1e-8f);
  float ca = c / det, cb = -b / det, cc = a / det;

  float op = sigmoidf(opacity[gid]);
  if (z <= ZNEAR) op = 0.f;   // pre-fold the z cull: alpha -> 0 -> log1m = 0
  float cr = sigmoidf(rgb[gid * 3 + 0]);
  float cg = sigmoidf(rgb[gid * 3 + 1]);
  float cbl = sigmoidf(rgb[gid * 3 + 2]);

  float* r = rec + (size_t)gid * GSTRIDE;
  r[0] = u;  r[1] = v;  r[2] = ca; r[3] = cb;
  r[4] = cc; r[5] = op; r[6] = cr; r[7] = cg;
  r[8] = cbl; r[9] = z; r[10] = 0.f; r[11] = 0.f;
  zkey[gid] = z;
}

// ---------------------------------------------------------------------------
// Kernel 2: bitonic depth sort (ascending z) of 2048 (key,idx) pairs in LDS,
// then gather unsorted records into sorted order. Single block, 1024 threads.
// ---------------------------------------------------------------------------
__global__ __launch_bounds__(1024) void gs_sort(const float* __restrict__ zkey,
                                                const float* __restrict__ rec_in,
                                                float* __restrict__ rec_out)
{
  __shared__ float skey[N_GAUSS];
  __shared__ int   sval[N_GAUSS];
  const int tid = threadIdx.x;

  for (int i = tid; i < N_GAUSS; i += 1024) { skey[i] = zkey[i]; sval[i] = i; }

  for (int k = 2; k <= N_GAUSS; k <<= 1) {
    for (int j = k >> 1; j > 0; j >>= 1) {
      __syncthreads();
      for (int i = tid; i < N_GAUSS; i += 1024) {
        int ixj = i ^ j;
        if (ixj > i) {
          bool up = ((i & k) == 0);
          float ki = skey[i], kj = skey[ixj];
          if ((ki > kj) == up) {
            skey[i] = kj; skey[ixj] = ki;
            int vi = sval[i]; sval[i] = sval[ixj]; sval[ixj] = vi;
          }
        }
      }
    }
  }
  __syncthreads();

  for (int i = tid; i < N_GAUSS; i += 1024) {
    const float4* src = (const float4*)(rec_in + (size_t)sval[i] * GSTRIDE);
    float4* dst = (float4*)(rec_out + (size_t)i * GSTRIDE);
    dst[0] = src[0]; dst[1] = src[1]; dst[2] = src[2];
  }
}

// ---------------------------------------------------------------------------
// Kernel 3: alpha compositing + WMMA color accumulation.
// One wave = one 16-pixel tile. Gaussians consumed front-to-back in chunks of
// K=4: lanes 0-15 own K={0,1}, lanes 16-31 own K={2,3} (ISA 16x4 f32 A layout).
// Transmittance scan crosses the half-waves via one shfl_xor(16) per chunk.
// D = A(16x4 weights) x B(4x16 colors, cols 3..15 = 0) accumulated with
// v_wmma_f32_16x16x4_f32 (512 WMMAs per wave).
// ---------------------------------------------------------------------------
__global__ __launch_bounds__(256) void gs_render(const float* __restrict__ gs,
                                                 const float* __restrict__ bgp,
                                                 float* __restrict__ out)
{
  const int lane = threadIdx.x & 31;
  const int wid  = threadIdx.x >> 5;
  const int tile = blockIdx.x * 8 + wid;     // 72 blocks * 8 waves = 576 tiles
  const int pixBase = tile * 16;
  const int m = lane & 15;                   // pixel-in-tile for alpha math
  const int kk = lane >> 4;                  // 0 -> K pair (0,1); 1 -> (2,3)
  const int p = pixBase + m;
  const float px = (float)(p % IMG) + 0.5f;
  const float py = (float)(p / IMG) + 0.5f;

  v8f acc = {};          // 16x16 f32 accumulator; columns 0..2 = RGB
  float S = 0.f;         // running log-transmittance for pixel m

  for (int g0 = 0; g0 < N_GAUSS; g0 += 4) {
    const int gA = g0 + 2 * kk;
    const float4* base = (const float4*)(gs + (size_t)gA * GSTRIDE);
    float4 A0 = base[0], A1 = base[1], A2 = base[2];   // gaussian gA
    float4 B0 = base[3], B1 = base[4], B2 = base[5];   // gaussian gA+1
    (void)A2; (void)B2;

    // alpha for gA: A0=[u,v,ca,cb] A1=[cc,op,r,g] A2=[b,z,0,0]
    float dxA = px - A0.x, dyA = py - A0.y;
    float powA = fminf(-0.5f * (A0.z * dxA * dxA + A1.x * dyA * dyA) - A0.w * dxA * dyA, 0.f);
    float alA = fminf(A1.y * __expf(powA), ALPHA_MAX);
    alA = (alA < ALPHA_MIN) ? 0.f : alA;
    float lA = __logf(1.f - alA);

    float dxB = px - B0.x, dyB = py - B0.y;
    float powB = fminf(-0.5f * (B0.z * dxB * dxB + B1.x * dyB * dyB) - B0.w * dxB * dyB, 0.f);
    float alB = fminf(B1.y * __expf(powB), ALPHA_MAX);
    alB = (alB < ALPHA_MIN) ? 0.f : alB;
    float lB = __logf(1.f - alB);

    // cross-half-wave transmittance scan (pixel m is mirrored in both halves)
    float hsum  = lA + lB;
    float other = __shfl_xor(hsum, 16, 32);          // low half gets (l2+l3), high gets (l0+l1)
    float Sstart = S + (kk ? other : 0.f);           // prefix for this lane's K pair
    float wA = alA * __expf(Sstart);
    float wB = alB * __expf(Sstart + lA);
    S = S + hsum + other;                            // identical in both halves

    // A operand: weights  (lane m, K = 2*kk / 2*kk+1)
    v2f av; av.x = wA; av.y = wB;
    // B operand: colors for this lane's K pair at column n = lane&15 (0..2 = RGB)
    float bx = (m == 0) ? A1.z : ((m == 1) ? A1.w : ((m == 2) ? A2.x : 0.f));
    float by = (m == 0) ? B1.z : ((m == 1) ? B1.w : ((m == 2) ? B2.x : 0.f));
    v2f bv; bv.x = bx; bv.y = by;

    acc = __builtin_amdgcn_wmma_f32_16x16x4_f32(
        /*neg_a=*/false, av, /*neg_b=*/false, bv,
        /*c_mod=*/(short)0, acc, /*reuse_a=*/false, /*reuse_b=*/false);
  }

  // D layout: VGPR i -> row M = i + (lane<16 ? 0 : 8), column N = lane&15.
  const int mbase = kk ? 8 : 0;
  const int n = m;                                  // output channel for this lane
  float bgv = (n < 3) ? bgp[n] : 0.f;
  float tf[8];
#pragma unroll
  for (int i = 0; i < 8; ++i)                       // pixel M's S lives in lane M
    tf[i] = __expf(__shfl(S, mbase + i, 32));
#pragma unroll
  for (int i = 0; i < 8; ++i) {
    if (n < 3)
      out[(size_t)(pixBase + mbase + i) * 3 + n] = acc[i] + tf[i] * bgv;
  }
}

// ---------------------------------------------------------------------------
extern "C" void kernel_launch(void* const* d_in, const int* in_sizes, int n_in,
                              void* d_out, int out_size, void* d_ws, size_t ws_size,
                              hipStream_t stream) {
  const float* xyz      = (const float*)d_in[0];
  const float* scaling  = (const float*)d_in[1];
  const float* rotation = (const float*)d_in[2];
  const float* opacity  = (const float*)d_in[3];
  const float* rgb      = (const float*)d_in[4];
  const float* c2w      = (const float*)d_in[5];
  const float* intr     = (const float*)d_in[6];
  const float* bg       = (const float*)d_in[7];
  float* out = (float*)d_out;

  // workspace: [ zkey(2048) | unsorted recs(2048*12) | sorted recs(2048*12) ] floats
  float* w = (float*)d_ws;
  float* zkey   = w;
  float* rec_u  = w + N_GAUSS;
  float* rec_s  = rec_u + (size_t)N_GAUSS * GSTRIDE;

  gs_preprocess<<<N_GAUSS / 256, 256, 0, stream>>>(xyz, scaling, rotation, opacity,
                                                   rgb, c2w, intr, rec_u, zkey);
  gs_sort<<<1, 1024, 0, stream>>>(zkey, rec_u, rec_s);
  gs_render<<<(NPIX / 16) / 8, 256, 0, stream>>>(rec_s, bg, out);
}